// ItemCodeLayer_30253749633338
// MI455X (gfx1250) — compile-verified
//
#include <hip/hip_runtime.h>
#include <stdint.h>

// ---------------- problem constants (match reference) ----------------
#define BB 512
#define SS 200
#define MM 8            // pq_m code bytes per item
#define SUBV 32         // floats per sub-embedding
#define NC 256          // centroids per code byte
#define ITEMS (BB * SS) // 102400 (b,s) pairs

// ---------------- tiling ----------------
#define THREADS 256
#define ITEMS_PER_BLOCK 1024               // 102400 / 1024 = 100 blocks in x
#define PASSES (ITEMS_PER_BLOCK / 32)      // 32 item-groups of 32 per pass
#define LDS_ROW 36                         // 32 data floats + 4 pad floats (TDM pad)

typedef __attribute__((ext_vector_type(4))) unsigned int u32x4;
typedef __attribute__((ext_vector_type(8))) int          i32x8;
typedef __attribute__((ext_vector_type(4))) int          i32x4;

// PQ decode: out[b,s, m*32 + j] = centroids[m, codes[ids[b,s]*8 + m], j]
// grid = (ITEMS/ITEMS_PER_BLOCK, MM); block = 256 threads (8 wave32).
// One TDM descriptor stages codebook m (8192 f32, 32 KB) into LDS with
// 16B padding every 128B so centroid rows stride 144B across the 64 banks.
__global__ __launch_bounds__(THREADS)
void ItemCodeLayer_pq_decode_kernel(const int*   __restrict__ ids,
                                    const int*   __restrict__ codes,
                                    const float* __restrict__ centroids,
                                    float*       __restrict__ out)
{
    __shared__ __align__(16) float lds[NC * LDS_ROW]; // 36 KB

    const int m = blockIdx.y;

    // -------- stage codebook m into LDS via Tensor Data Mover --------
    if (threadIdx.x == 0) {
        const uint64_t gaddr = (uint64_t)(uintptr_t)(centroids + (size_t)m * NC * SUBV);
        const uint32_t laddr = (uint32_t)(uintptr_t)(void*)lds; // flat LDS addr: [31:0] = offset

        u32x4 g0;
        g0.x = 1u;                                   // count=1 (valid user descriptor)
        g0.y = laddr;                                // lds_addr
        g0.z = (uint32_t)gaddr;                      // global_addr[31:0]
        g0.w = (uint32_t)((gaddr >> 32) & 0x01FFFFFFu) | (2u << 30); // addr[56:32] | type=2

        i32x8 g1;
        // data_size=4B (2<<16) | pad_enable (1<<20) | pad_interval: 32 dwords (4<<22)
        // | pad_amount: 4 dwords (3<<25); workgroup_mask=0 (not in a cluster)
        g1[0] = (int)((2u << 16) | (1u << 20) | (4u << 22) | (3u << 25));
        g1[1] = (int)((8192u & 0xFFFFu) << 16);      // tensor_dim0[15:0] at bits[63:48]
        g1[2] = (int)(1u << 16);                     // tensor_dim0 hi = 0; tensor_dim1 = 1
        g1[3] = (int)((8192u & 0xFFFFu) << 16);      // tile_dim0 = 8192 elements
        g1[4] = 1;                                   // tile_dim1 = 1, tile_dim2 = 0
        g1[5] = 8192;                                // tensor_dim0_stride[31:0]
        g1[6] = 0;                                   // stride0 hi / stride1 lo
        g1[7] = 0;

        i32x4 z4 = {0, 0, 0, 0};                     // groups 2/3 unused (<=2D tensor)
        i32x8 z8 = {0, 0, 0, 0, 0, 0, 0, 0};         // extra group (6-arg toolchain), unused
        __builtin_amdgcn_tensor_load_to_lds(g0, g1, z4, z4, z8, 0);
        __builtin_amdgcn_s_wait_tensorcnt(0);        // TENSORcnt == 0: tile resident in LDS
    }
    __syncthreads();

    // -------- gather: 8 threads per item, one float4 each --------
    const int q    = threadIdx.x & 7;                // float4 slot within sub-embedding
    const int ig   = threadIdx.x >> 3;               // item group 0..31
    const int base = blockIdx.x * ITEMS_PER_BLOCK;

    const float4* ldsv = (const float4*)lds;         // row stride = LDS_ROW/4 = 9 float4
    float4*       out4 = (float4*)out;

    #pragma unroll 4
    for (int p = 0; p < PASSES; ++p) {
        const int item = base + p * 32 + ig;
        const int id   = ids[item];                  // L2-resident after warmup
        const int code = codes[id * MM + m];         // 1 dword gather per (item,m)
        const float4 v = ldsv[code * (LDS_ROW / 4) + q];   // ds_load_b128, padded rows
        out4[(size_t)item * (MM * SUBV / 4) + m * (SUBV / 4) + q] = v; // coalesced 128B/item
    }
}

extern "C" void kernel_launch(void* const* d_in, const int* in_sizes, int n_in,
                              void* d_out, int out_size, void* d_ws, size_t ws_size,
                              hipStream_t stream) {
    (void)in_sizes; (void)n_in; (void)out_size; (void)d_ws; (void)ws_size;
    const int*   ids       = (const int*)d_in[0];   // input_ids  [B*S]
    const int*   codes     = (const int*)d_in[1];   // item_codes [(NUM_ITEMS+1)*M]
    const float* centroids = (const float*)d_in[2]; // centroids  [M*256*32]
    float*       out       = (float*)d_out;         // [B*S*256] f32

    dim3 grid(ITEMS / ITEMS_PER_BLOCK, MM, 1);      // (100, 8)
    dim3 block(THREADS, 1, 1);
    ItemCodeLayer_pq_decode_kernel<<<grid, block, 0, stream>>>(ids, codes, centroids, out);
}